// Model_24232205484603
// MI455X (gfx1250) — compile-verified
//
#include <hip/hip_runtime.h>

// Deblocking filter, 8192x8192 fp32, BS=8, ALPHA=0.1, BETA=0.05.
// Pure HBM-bandwidth problem (~0.02 flop/byte) -> no WMMA; optimize the
// data-movement path: b128 vmem, CDNA5 temporal hints (NT vs RT) chosen so
// the 128MB of rows re-read by the horizontal pass stays resident in the
// 192MB L2, plus WGP-scope global_prefetch for the dependent halo loads.

typedef float v4f __attribute__((ext_vector_type(4)));
typedef float v2f __attribute__((ext_vector_type(2)));

#define FW 8192
#define FH 8192
#define ALPHA 0.1f
#define BETA  0.05f

__device__ __forceinline__ float deblk(float a, float b, float c) {
    // (2*a + b + c + 2) / 4  -- matches reference evaluation order
    return (2.0f * a + b + c + 2.0f) * 0.25f;
}

// ---------------------------------------------------------------------------
// Pass 1: vertical (column-boundary) filter, fused with the frame copy.
// One thread = one aligned 8-column group of one row.
//   boundary c = 8*blk   : updates q0 = col 8*blk      (needs left halo)
//   boundary c = 8*blk+8 : updates p0 = col 8*blk+7    (needs right halo)
// Input loads: NT (read-once, keep out of L2).
// Output stores: RT for rows the horizontal pass re-reads (row%8 in {6,7,0,1},
// 128MB total -> fits MI455X's 192MB L2), NT streaming for the rest.
// ---------------------------------------------------------------------------
__global__ __launch_bounds__(256) void vpass_kernel(const float* __restrict__ in,
                                                    float* __restrict__ out) {
    const int  row  = blockIdx.y;
    const int  blk  = blockIdx.x * blockDim.x + threadIdx.x;     // 0..1023
    const long base = (long)row * FW + (long)blk * 8;
    const float* p  = in + base;

    // Warm the right-halo cacheline early. Locality 3 -> WGP scope: pulls the
    // line through all cache levels for this wave's own dependent load
    // (SYS-scope prefetch would stop at GL2).
    __builtin_prefetch(p + 8, 0, 3);

    v4f a = __builtin_nontemporal_load((const v4f*)p);           // cols +0..+3
    v4f b = __builtin_nontemporal_load((const v4f*)(p + 4));     // cols +4..+7

    if (blk > 0) {                       // boundary c = 8*blk in [8, 8184]
        v2f l = __builtin_nontemporal_load((const v2f*)(p - 2)); // {p1, p0}
        const float p1 = l.x, p0 = l.y, q0 = a.x, q1 = a.y;
        const bool m = (fabsf(p0 - q0) < ALPHA) &
                       (fabsf(p1 - p0) < BETA)  &
                       (fabsf(q1 - q0) < BETA);
        if (m) a.x = deblk(q1, q0, p0);  // q0_new
    }
    if (blk < 1023) {                    // boundary c = 8*blk + 8 in [8, 8184]
        v2f r = __builtin_nontemporal_load((const v2f*)(p + 8)); // {q0, q1}
        const float p1 = b.z, p0 = b.w, q0 = r.x, q1 = r.y;
        const bool m = (fabsf(p0 - q0) < ALPHA) &
                       (fabsf(p1 - p0) < BETA)  &
                       (fabsf(q1 - q0) < BETA);
        if (m) b.w = deblk(p1, p0, q0);  // p0_new
    }

    float* q = out + base;
    const int  rm     = row & 7;
    const bool reread = (rm <= 1) | (rm >= 6);   // rows pass 2 reads again
    if (reread) {                                // keep in L2 for hpass
        *(v4f*)q       = a;
        *(v4f*)(q + 4) = b;
    } else {                                     // stream straight to HBM
        __builtin_nontemporal_store(a, (v4f*)q);
        __builtin_nontemporal_store(b, (v4f*)(q + 4));
    }
}

// ---------------------------------------------------------------------------
// Pass 2: horizontal (row-boundary) filter, in place on d_out.
// One thread = 4 columns of one boundary row r in {8,16,...,8184}.
// Hazard-free in place: written rows (r-1, r) are disjoint across boundaries
// and never read by other threads; each thread reads before it writes.
// NT loads: hit the lines pass 1 parked in L2, then release them.
// NT stores: final output, never re-read on device.
// ---------------------------------------------------------------------------
__global__ __launch_bounds__(256) void hpass_kernel(float* __restrict__ out) {
    const int  r = (blockIdx.y + 1) * 8;                         // 8..8184
    const long c = ((long)blockIdx.x * blockDim.x + threadIdx.x) * 4;
    float* colp  = out + c;

    v4f p1 = __builtin_nontemporal_load((const v4f*)(colp + (long)(r - 2) * FW));
    v4f p0 = __builtin_nontemporal_load((const v4f*)(colp + (long)(r - 1) * FW));
    v4f q0 = __builtin_nontemporal_load((const v4f*)(colp + (long)(r    ) * FW));
    v4f q1 = __builtin_nontemporal_load((const v4f*)(colp + (long)(r + 1) * FW));

    v4f p0n = p0, q0n = q0;
#pragma unroll
    for (int i = 0; i < 4; ++i) {
        const bool m = (fabsf(p0[i] - q0[i]) < ALPHA) &
                       (fabsf(p1[i] - p0[i]) < BETA)  &
                       (fabsf(q1[i] - q0[i]) < BETA);
        if (m) {
            p0n[i] = deblk(p1[i], p0[i], q0[i]);
            q0n[i] = deblk(q1[i], q0[i], p0[i]);
        }
    }

    __builtin_nontemporal_store(p0n, (v4f*)(colp + (long)(r - 1) * FW));
    __builtin_nontemporal_store(q0n, (v4f*)(colp + (long)(r    ) * FW));
}

extern "C" void kernel_launch(void* const* d_in, const int* in_sizes, int n_in,
                              void* d_out, int out_size, void* d_ws, size_t ws_size,
                              hipStream_t stream) {
    const float* frame = (const float*)d_in[0];
    float*       out   = (float*)d_out;

    // Pass 1: 1024 threads/row (8 cols each) x 8192 rows.
    vpass_kernel<<<dim3(4, FH), dim3(256), 0, stream>>>(frame, out);
    // Pass 2: 2048 col-groups x 1023 boundary rows, in place on out.
    hpass_kernel<<<dim3(8, 1023), dim3(256), 0, stream>>>(out);
}